// SearchSPOSGCNConv_14370960573135
// MI455X (gfx1250) — compile-verified
//
#include <hip/hip_runtime.h>
#include <hip/hip_bf16.h>

typedef __attribute__((ext_vector_type(2))) float v2f;
typedef __attribute__((ext_vector_type(8))) float v8f;

#define V_NODES 100000
#define N_EDGES 524288
#define DIM 128
#define BN_EPS_F 1e-5f
#define GEMM_BLOCKS 256

// ---------------------------------------------------------------------------
// Kernel 1: per-edge gather, comp('mult'), scale by edge_norm, scatter-add
// into pre-GEMM accumulators (acc_in for first half of edges, acc_out for
// second half).  One wave per edge; lane handles 4 channels (float4).
// x (51 MB) and rel (243 KB) are L2-resident on MI455X (192 MB L2).
// ---------------------------------------------------------------------------
__global__ __launch_bounds__(256) void edge_scatter_kernel(
    const float* __restrict__ x, const float* __restrict__ rel,
    const float* __restrict__ edge_norm, const int* __restrict__ src,
    const int* __restrict__ dst, const int* __restrict__ etype,
    float* __restrict__ acc_in, float* __restrict__ acc_out,
    int E, int halfE)
{
    const int e    = blockIdx.x * (blockDim.x >> 5) + (threadIdx.x >> 5);
    const int lane = threadIdx.x & 31;
    if (e >= E) return;

    const int   s   = src[e];
    const int   d   = dst[e];
    const int   t   = etype[e];
    const float nrm = edge_norm[e];

    const float4 xv = *(const float4*)(x   + (size_t)s * DIM + lane * 4);
    const float4 rv = *(const float4*)(rel + (size_t)t * DIM + lane * 4);

    float* acc = (e < halfE) ? acc_in : acc_out;
    float* p   = acc + (size_t)d * DIM + lane * 4;

    atomicAdd(p + 0, nrm * xv.x * rv.x);
    atomicAdd(p + 1, nrm * xv.y * rv.y);
    atomicAdd(p + 2, nrm * xv.z * rv.z);
    atomicAdd(p + 3, nrm * xv.w * rv.w);
}

// ---------------------------------------------------------------------------
// Kernel 2: node-level fused GEMM using V_WMMA_F32_16X16X4_F32.
//   h = (acc_in @ in_w + acc_out @ out_w + (x*loop_rel) @ loop_w) / 3 + bias
// Persistent grid-stride kernel: each block stages all three 128x128 weights
// in LDS once (192 KB), then loops over 32-row units.  Each wave owns TWO
// 16x128 output tiles so every B fragment (2 ds_load_b32) feeds 2 WMMAs
// (48 wmma : 48 ds_load per K-step -> matrix-unit bound, not LDS bound).
// Also accumulates per-channel sum/sumsq for BatchNorm via LDS ds_add_f32,
// flushed with one global atomic per channel per block at the end.
// ---------------------------------------------------------------------------
__global__ __launch_bounds__(256) void node_gemm_kernel(
    const float* __restrict__ acc_in, const float* __restrict__ acc_out,
    const float* __restrict__ x, const float* __restrict__ loop_rel,
    const float* __restrict__ in_w, const float* __restrict__ out_w,
    const float* __restrict__ loop_w, const float* __restrict__ bias,
    float* __restrict__ h_out, float* __restrict__ gsum,
    float* __restrict__ gsumsq, int num_units)
{
    extern __shared__ float smem[];
    float* sw   = smem;                 // 3 * 128 * 128 floats
    float* ssum = smem + 3 * DIM * DIM; // 256 floats: [0..127]=sum [128..255]=sumsq

    const int tid = threadIdx.x;

    // Cooperative load of the three weight matrices into LDS (float4 copies).
    {
        const float* wsrc[3] = { in_w, out_w, loop_w };
        for (int m = 0; m < 3; ++m) {
            const float4* srcp = (const float4*)wsrc[m];
            float4*       dstp = (float4*)(sw + m * DIM * DIM);
            for (int i = tid; i < (DIM * DIM) / 4; i += blockDim.x)
                dstp[i] = srcp[i];
        }
        ssum[tid] = 0.0f;   // blockDim == 256 covers all 256 slots
    }
    __syncthreads();

    const int wave  = tid >> 5;
    const int lane  = tid & 31;
    const int l16   = lane & 15;
    const int lhalf = lane >> 4;               // 0: lanes 0-15, 1: lanes 16-31

    for (int unit = blockIdx.x * 8 + wave; unit < num_units;
         unit += (int)gridDim.x * 8) {
        const int    m0     = unit * 32;       // 32 rows per unit (2 M-tiles)
        const size_t abase0 = (size_t)(m0 + l16) * DIM;
        const size_t abase1 = (size_t)(m0 + 16 + l16) * DIM;

        v8f c0[8], c1[8];
        #pragma unroll
        for (int nt = 0; nt < 8; ++nt)
            #pragma unroll
            for (int i = 0; i < 8; ++i) { c0[nt][i] = 0.0f; c1[nt][i] = 0.0f; }

        for (int kt = 0; kt < DIM / 4; ++kt) {
            const int k0 = kt * 4 + lhalf * 2; // this half-wave's K columns

            // A fragments (16x4 f32): lane<16 -> K={k,k+1}, lane>=16 -> K={k+2,k+3}
            v2f a_in0 = *(const v2f*)(acc_in  + abase0 + k0);
            v2f a_ot0 = *(const v2f*)(acc_out + abase0 + k0);
            v2f a_lp0 = *(const v2f*)(x       + abase0 + k0);
            v2f a_in1 = *(const v2f*)(acc_in  + abase1 + k0);
            v2f a_ot1 = *(const v2f*)(acc_out + abase1 + k0);
            v2f a_lp1 = *(const v2f*)(x       + abase1 + k0);
            const float lr0 = loop_rel[k0];
            const float lr1 = loop_rel[k0 + 1];
            a_lp0.x *= lr0; a_lp0.y *= lr1;
            a_lp1.x *= lr0; a_lp1.y *= lr1;

            #pragma unroll
            for (int nt = 0; nt < 8; ++nt) {
                const int n = nt * 16 + l16;
                v2f b;
                // B fragment (4x16): b.x = W[k0][n], b.y = W[k0+1][n]
                b.x = sw[0 * DIM * DIM + k0 * DIM + n];
                b.y = sw[0 * DIM * DIM + (k0 + 1) * DIM + n];
                c0[nt] = __builtin_amdgcn_wmma_f32_16x16x4_f32(
                             false, a_in0, false, b, (short)0, c0[nt], false, false);
                c1[nt] = __builtin_amdgcn_wmma_f32_16x16x4_f32(
                             false, a_in1, false, b, (short)0, c1[nt], false, false);
                b.x = sw[1 * DIM * DIM + k0 * DIM + n];
                b.y = sw[1 * DIM * DIM + (k0 + 1) * DIM + n];
                c0[nt] = __builtin_amdgcn_wmma_f32_16x16x4_f32(
                             false, a_ot0, false, b, (short)0, c0[nt], false, false);
                c1[nt] = __builtin_amdgcn_wmma_f32_16x16x4_f32(
                             false, a_ot1, false, b, (short)0, c1[nt], false, false);
                b.x = sw[2 * DIM * DIM + k0 * DIM + n];
                b.y = sw[2 * DIM * DIM + (k0 + 1) * DIM + n];
                c0[nt] = __builtin_amdgcn_wmma_f32_16x16x4_f32(
                             false, a_lp0, false, b, (short)0, c0[nt], false, false);
                c1[nt] = __builtin_amdgcn_wmma_f32_16x16x4_f32(
                             false, a_lp1, false, b, (short)0, c1[nt], false, false);
            }
        }

        // Epilogue: scale 1/3, add bias, store, accumulate BN partial stats.
        // C/D layout: VGPR r, lanes 0-15 -> row m0+r; lanes 16-31 -> row m0+8+r.
        const int rbase0 = m0 + lhalf * 8;
        const int rbase1 = m0 + 16 + lhalf * 8;
        #pragma unroll
        for (int nt = 0; nt < 8; ++nt) {
            const int   n  = nt * 16 + l16;
            const float bz = bias[n];
            float s = 0.0f, q = 0.0f;
            #pragma unroll
            for (int r = 0; r < 8; ++r) {
                float h0 = c0[nt][r] * (1.0f / 3.0f) + bz;
                float h1 = c1[nt][r] * (1.0f / 3.0f) + bz;
                h_out[(size_t)(rbase0 + r) * DIM + n] = h0;
                h_out[(size_t)(rbase1 + r) * DIM + n] = h1;
                s += h0 + h1;
                q += h0 * h0 + h1 * h1;
            }
            // lanes L and L+16 cover the same channel n: combine, then one
            // LDS atomic per channel from the low half-wave.
            s += __shfl_xor(s, 16, 32);
            q += __shfl_xor(q, 16, 32);
            if (lhalf == 0) {
                atomicAdd(&ssum[n], s);
                atomicAdd(&ssum[DIM + n], q);
            }
        }
    }
    __syncthreads();
    if (tid < DIM)          atomicAdd(&gsum[tid],          ssum[tid]);
    else if (tid < 2 * DIM) atomicAdd(&gsumsq[tid - DIM],  ssum[tid]);
}

// ---------------------------------------------------------------------------
// Kernel 3: convert channel sums -> mean / inv_std in place.
// ---------------------------------------------------------------------------
__global__ void bn_stats_kernel(float* __restrict__ gsum,
                                float* __restrict__ gsumsq, float invV)
{
    const int c    = threadIdx.x;
    const float m  = gsum[c] * invV;
    const float vr = gsumsq[c] * invV - m * m;
    gsum[c]   = m;
    gsumsq[c] = rsqrtf(vr + BN_EPS_F);
}

// ---------------------------------------------------------------------------
// Kernel 4: in-place BatchNorm + ReLU over h (float4 per thread).
// ---------------------------------------------------------------------------
__global__ __launch_bounds__(256) void bn_relu_kernel(
    float* __restrict__ h, const float* __restrict__ mean,
    const float* __restrict__ istd, long long n4)
{
    const long long i = (long long)blockIdx.x * blockDim.x + threadIdx.x;
    if (i >= n4) return;
    const int c = (int)((i * 4) & (DIM - 1));
    float4 v = ((float4*)h)[i];
    v.x = fmaxf(0.0f, (v.x - mean[c + 0]) * istd[c + 0]);
    v.y = fmaxf(0.0f, (v.y - mean[c + 1]) * istd[c + 1]);
    v.z = fmaxf(0.0f, (v.z - mean[c + 2]) * istd[c + 2]);
    v.w = fmaxf(0.0f, (v.w - mean[c + 3]) * istd[c + 3]);
    ((float4*)h)[i] = v;
}

// ---------------------------------------------------------------------------
// Kernel 5: rel_repr [474,128] @ w_rel [128,128] via WMMA, row-guarded.
// One wave (32 threads) per 16-row tile; w_rel is small and L1/L2 resident.
// ---------------------------------------------------------------------------
__global__ __launch_bounds__(32) void rel_gemm_kernel(
    const float* __restrict__ rel, const float* __restrict__ w_rel,
    float* __restrict__ out, int M)
{
    const int lane  = threadIdx.x & 31;
    const int l16   = lane & 15;
    const int lhalf = lane >> 4;
    const int m0    = blockIdx.x * 16;
    const int arow  = m0 + l16;
    const bool rowok = (arow < M);

    v8f c[8];
    #pragma unroll
    for (int nt = 0; nt < 8; ++nt)
        #pragma unroll
        for (int i = 0; i < 8; ++i) c[nt][i] = 0.0f;

    for (int kt = 0; kt < DIM / 4; ++kt) {
        const int k0 = kt * 4 + lhalf * 2;
        v2f a;
        if (rowok) {
            a = *(const v2f*)(rel + (size_t)arow * DIM + k0);
        } else {
            a.x = 0.0f; a.y = 0.0f;
        }
        #pragma unroll
        for (int nt = 0; nt < 8; ++nt) {
            const int n = nt * 16 + l16;
            v2f b;
            b.x = w_rel[k0 * DIM + n];
            b.y = w_rel[(k0 + 1) * DIM + n];
            c[nt] = __builtin_amdgcn_wmma_f32_16x16x4_f32(
                        false, a, false, b, (short)0, c[nt], false, false);
        }
    }

    const int rbase = m0 + lhalf * 8;
    #pragma unroll
    for (int nt = 0; nt < 8; ++nt) {
        const int n = nt * 16 + l16;
        #pragma unroll
        for (int r = 0; r < 8; ++r) {
            const int row = rbase + r;
            if (row < M) out[(size_t)row * DIM + n] = c[nt][r];
        }
    }
}

// ---------------------------------------------------------------------------
extern "C" void kernel_launch(void* const* d_in, const int* in_sizes, int n_in,
                              void* d_out, int out_size, void* d_ws, size_t ws_size,
                              hipStream_t stream)
{
    (void)in_sizes; (void)n_in; (void)out_size; (void)ws_size;

    const float* x        = (const float*)d_in[0];
    const float* rel_repr = (const float*)d_in[1];
    const float* edge_nrm = (const float*)d_in[2];
    const float* in_w     = (const float*)d_in[3];
    const float* out_w    = (const float*)d_in[4];
    const float* loop_w   = (const float*)d_in[5];
    const float* w_rel    = (const float*)d_in[6];
    const float* loop_rel = (const float*)d_in[7];
    const float* bias     = (const float*)d_in[8];
    const int*   src      = (const int*)d_in[9];
    const int*   dst      = (const int*)d_in[10];
    const int*   etype    = (const int*)d_in[11];

    float* out = (float*)d_out;

    // Workspace layout: acc_in[V*128] | acc_out[V*128] | gsum[128] | gsumsq[128]
    float* acc_in  = (float*)d_ws;
    float* acc_out = acc_in  + (size_t)V_NODES * DIM;
    float* gsum    = acc_out + (size_t)V_NODES * DIM;
    float* gsumsq  = gsum + DIM;

    const size_t zero_bytes = ((size_t)2 * V_NODES * DIM + 2 * DIM) * sizeof(float);
    hipMemsetAsync(d_ws, 0, zero_bytes, stream);

    // 1) edge scatter into pre-GEMM accumulators
    edge_scatter_kernel<<<N_EDGES / 8, 256, 0, stream>>>(
        x, rel_repr, edge_nrm, src, dst, etype,
        acc_in, acc_out, N_EDGES, N_EDGES / 2);

    // 2) fused node GEMM (WMMA), persistent blocks, + BN statistics
    const int num_units = V_NODES / 32;            // 3125 units of 32 rows
    const size_t smem   = (3 * DIM * DIM + 2 * DIM) * sizeof(float); // ~193 KB
    node_gemm_kernel<<<GEMM_BLOCKS, 256, smem, stream>>>(
        acc_in, acc_out, x, loop_rel, in_w, out_w, loop_w, bias,
        out, gsum, gsumsq, num_units);

    // 3) BN stats finalize, 4) BN + ReLU in place
    bn_stats_kernel<<<1, DIM, 0, stream>>>(gsum, gsumsq, 1.0f / (float)V_NODES);
    const long long n4 = (long long)V_NODES * DIM / 4;
    bn_relu_kernel<<<(int)((n4 + 255) / 256), 256, 0, stream>>>(out, gsum, gsumsq, n4);

    // 5) relation projection (independent small WMMA GEMM)
    rel_gemm_kernel<<<(474 + 15) / 16, 32, 0, stream>>>(
        rel_repr, w_rel, out + (size_t)V_NODES * DIM, 474);
}